// ClearMeshLoss_65369402245254
// MI455X (gfx1250) — compile-verified
//
#include <hip/hip_runtime.h>
#include <math.h>

// ---------------------------------------------------------------------------
// Mesh loss for MI455X (gfx1250, wave32). Chamfer/normal pairwise-distance
// dot products run on V_WMMA_F32_16X16X4_F32 (K=4, xyz + zero pad).
// Nearest-neighbor min/argmin computed in two symmetric passes (P->G, G->P)
// so the hot loop has no shuffles and no atomics: plain stores only.
// ---------------------------------------------------------------------------

typedef float v2f __attribute__((ext_vector_type(2)));
typedef float v8f __attribute__((ext_vector_type(8)));

#define TRUNC_ 0.1f
#define SURF_W_ 5.0f
#define DIH_THR_ 0.5f
#define EMPTY64 0xFFFFFFFFFFFFFFFFull

// float -> order-preserving unsigned key (total order, smaller float = smaller key)
__device__ __forceinline__ unsigned f2ord(float f) {
  unsigned u = __float_as_uint(f);
  return (u & 0x80000000u) ? ~u : (u | 0x80000000u);
}
__device__ __forceinline__ float ord2f(unsigned k) {
  unsigned u = (k & 0x80000000u) ? (k ^ 0x80000000u) : ~k;
  return __uint_as_float(u);
}

__device__ __forceinline__ unsigned hash_key(unsigned long long k, unsigned mask) {
  k *= 0x9E3779B97F4A7C15ull;
  k ^= k >> 32;
  return (unsigned)k & mask;
}

// block-wide sum, blockDim.x == 256
__device__ float blk_sum(float v, float* sh) {
  int t = threadIdx.x;
  __syncthreads();
  sh[t] = v;
  __syncthreads();
  for (int s = 128; s > 0; s >>= 1) {
    if (t < s) sh[t] += sh[t + s];
    __syncthreads();
  }
  return sh[0];
}

// ---------------------------------------------------------------------------
__global__ void k_init(unsigned long long* hkeys, unsigned* hcnt, int cap) {
  int stride = blockDim.x * gridDim.x;
  for (int i = blockIdx.x * blockDim.x + threadIdx.x; i < cap; i += stride) {
    hkeys[i] = EMPTY64;
    hcnt[i] = 0u;
  }
}

// squared norms of point sets
__global__ void k_norms(const float* __restrict__ P, const float* __restrict__ G,
                        float* __restrict__ a2, float* __restrict__ b2, int npts) {
  int stride = blockDim.x * gridDim.x;
  for (int i = blockIdx.x * blockDim.x + threadIdx.x; i < npts; i += stride) {
    float x = P[3 * i], y = P[3 * i + 1], z = P[3 * i + 2];
    a2[i] = x * x + y * y + z * z;
    x = G[3 * i]; y = G[3 * i + 1]; z = G[3 * i + 2];
    b2[i] = x * x + y * y + z * z;
  }
}

// SDF truncated-weighted-L1 + eikonal finite difference
__global__ void k_sdf_eik(const float* __restrict__ pred, const float* __restrict__ gt,
                          int B, int nsdf,
                          float* __restrict__ psdf, float* __restrict__ peik,
                          float* __restrict__ pcnt) {
  __shared__ float sh[256];
  int stride = blockDim.x * gridDim.x;
  int gtid = blockIdx.x * blockDim.x + threadIdx.x;
  float ssum = 0.f, esum = 0.f, ecnt = 0.f;
  int N = B * nsdf;
  for (int i = gtid; i < N; i += stride) {
    float g = gt[i], p = pred[i];
    float gc = fminf(fmaxf(g, -TRUNC_), TRUNC_);
    float pc = fminf(fmaxf(p, -TRUNC_), TRUNC_);
    float w = 1.f + (SURF_W_ - 1.f) * expf(-fabsf(gc) * (3.0f / TRUNC_));
    ssum += w * fabsf(pc - gc);
  }
  int Ne = B * (nsdf - 1);
  for (int j = gtid; j < Ne; j += stride) {
    int b = j / (nsdf - 1);
    int r = j - b * (nsdf - 1);
    int base = b * nsdf + r;
    if (fabsf(gt[base]) < TRUNC_) {
      float t = fabsf(pred[base + 1] - pred[base]) - 1.f;
      esum += t * t;
      ecnt += 1.f;
    }
  }
  float r0 = blk_sum(ssum, sh);
  float r1 = blk_sum(esum, sh);
  float r2 = blk_sum(ecnt, sh);
  if (threadIdx.x == 0) { psdf[blockIdx.x] = r0; peik[blockIdx.x] = r1; pcnt[blockIdx.x] = r2; }
}

// unit face normals
__global__ void k_facenorm(const float* __restrict__ verts, const int* __restrict__ faces,
                           float* __restrict__ fn, int Fn) {
  int stride = blockDim.x * gridDim.x;
  for (int f = blockIdx.x * blockDim.x + threadIdx.x; f < Fn; f += stride) {
    int i0 = faces[3 * f], i1 = faces[3 * f + 1], i2 = faces[3 * f + 2];
    float ax = verts[3 * i0], ay = verts[3 * i0 + 1], az = verts[3 * i0 + 2];
    float e1x = verts[3 * i1] - ax, e1y = verts[3 * i1 + 1] - ay, e1z = verts[3 * i1 + 2] - az;
    float e2x = verts[3 * i2] - ax, e2y = verts[3 * i2 + 1] - ay, e2z = verts[3 * i2 + 2] - az;
    float cx = e1y * e2z - e1z * e2y;
    float cy = e1z * e2x - e1x * e2z;
    float cz = e1x * e2y - e1y * e2x;
    float inv = 1.0f / fmaxf(sqrtf(cx * cx + cy * cy + cz * cz), 1e-12f);
    fn[3 * f] = cx * inv; fn[3 * f + 1] = cy * inv; fn[3 * f + 2] = cz * inv;
  }
}

// hash-insert the 3*Fn undirected edges; record up to 2 incident faces
__global__ void k_edge_insert(const int* __restrict__ faces,
                              unsigned long long* __restrict__ hkeys,
                              unsigned* __restrict__ hcnt,
                              unsigned* __restrict__ hfa, unsigned* __restrict__ hfb,
                              int Fn, int V, unsigned mask) {
  int stride = blockDim.x * gridDim.x;
  int nE = 3 * Fn;
  for (int e = blockIdx.x * blockDim.x + threadIdx.x; e < nE; e += stride) {
    int f = e / 3, c = e - 3 * f;
    int a = faces[3 * f + c];
    int b = faces[3 * f + ((c + 1) % 3)];
    int lo = a < b ? a : b;
    int hi = a < b ? b : a;
    unsigned long long key = (unsigned long long)lo * (unsigned long long)V + (unsigned long long)hi;
    unsigned h = hash_key(key, mask);
    for (;;) {
      unsigned long long prev = atomicCAS(&hkeys[h], EMPTY64, key);
      if (prev == EMPTY64 || prev == key) {
        unsigned pos = atomicAdd(&hcnt[h], 1u);
        if (pos == 0u) hfa[h] = (unsigned)f;
        else if (pos == 1u) hfb[h] = (unsigned)f;
        break;
      }
      h = (h + 1) & mask;
    }
  }
}

// evaluate dihedral-edge loss terms and watertight counts from the hash table
__global__ void k_edge_eval(const unsigned long long* __restrict__ hkeys,
                            const unsigned* __restrict__ hcnt,
                            const unsigned* __restrict__ hfa, const unsigned* __restrict__ hfb,
                            const float* __restrict__ fn,
                            float* __restrict__ pedge, float* __restrict__ pnp,
                            float* __restrict__ ptot, float* __restrict__ pbad, int cap) {
  __shared__ float sh[256];
  int stride = blockDim.x * gridDim.x;
  float es = 0.f, np = 0.f, tot = 0.f, bad = 0.f;
  for (int i = blockIdx.x * blockDim.x + threadIdx.x; i < cap; i += stride) {
    if (hkeys[i] != EMPTY64) {
      tot += 1.f;
      unsigned c = hcnt[i];
      if (c == 2u) {
        np += 1.f;
        unsigned fa = hfa[i], fb = hfb[i];
        float d = fn[3 * fa] * fn[3 * fb] + fn[3 * fa + 1] * fn[3 * fb + 1] +
                  fn[3 * fa + 2] * fn[3 * fb + 2];
        es += fmaxf(d - DIH_THR_, 0.f);
      } else {
        bad += 1.f;
      }
    }
  }
  float r0 = blk_sum(es, sh);
  float r1 = blk_sum(np, sh);
  float r2 = blk_sum(tot, sh);
  float r3 = blk_sum(bad, sh);
  if (threadIdx.x == 0) { pedge[blockIdx.x] = r0; pnp[blockIdx.x] = r1;
                          ptot[blockIdx.x] = r2; pbad[blockIdx.x] = r3; }
}

// ---------------------------------------------------------------------------
// Nearest-neighbor row min+argmin via V_WMMA_F32_16X16X4_F32.
// One wave owns one 16-row tile of A and sweeps all 16-col tiles of B.
// d_ij = |a_i|^2 + |b_j|^2 - 2 * (A . B^T)_ij
// A-matrix (16x4): M = lane%16; lanes 0-15 carry K={0,1}=(x,y), lanes 16-31
// carry K={2,3}=(z,0). B-matrix (4x16) mirrors with N = lane%16.
// C layout: c[r] for lane l -> row = r + 8*(l>=16), col = l%16.
// Row tiles are exclusively owned -> results stored with plain (deterministic)
// stores; no atomics, no LDS in the hot loop. Called twice with roles swapped.
// ---------------------------------------------------------------------------
__global__ void k_nnmin(const float* __restrict__ A, const float* __restrict__ B,
                        const float* __restrict__ na2, const float* __restrict__ nb2,
                        unsigned long long* __restrict__ minArg,
                        int nA, int nB) {
  const int lane = threadIdx.x & 31;
  const int col16 = lane & 15;
  const int half = lane >> 4;
  const int wave = (blockIdx.x * blockDim.x + threadIdx.x) >> 5;
  const int nwaves = (gridDim.x * blockDim.x) >> 5;
  const int ntA = (nA + 15) / 16;
  const int ntB = (nB + 15) / 16;

  for (int rt = wave; rt < ntA; rt += nwaves) {
    const int rowm = rt * 16 + col16;  // A-matrix row for this lane
    float px = 0.f, py = 0.f, pz = 0.f;
    if (rowm < nA) { px = A[3 * rowm]; py = A[3 * rowm + 1]; pz = A[3 * rowm + 2]; }
    v2f a;
    a.x = half ? pz : px;
    a.y = half ? 0.f : py;

    const int rbase = rt * 16 + half * 8;  // rows covered by this lane's c[0..7]
    float a2r[8];
    unsigned long long rmin[8];
#pragma unroll
    for (int r = 0; r < 8; ++r) {
      int rr = rbase + r;
      a2r[r] = (rr < nA) ? na2[rr] : 3.0e38f;  // OOB rows -> huge distance
      rmin[r] = EMPTY64;
    }

#pragma unroll 2
    for (int ct = 0; ct < ntB; ++ct) {
      const int n = ct * 16 + col16;  // B-matrix column for this lane
      float gx = 0.f, gy = 0.f, gz = 0.f, b2n = 3.0e38f;
      if (n < nB) { gx = B[3 * n]; gy = B[3 * n + 1]; gz = B[3 * n + 2]; b2n = nb2[n]; }
      v2f bv;
      bv.x = half ? gz : gx;
      bv.y = half ? 0.f : gy;

      v8f c = {0.f, 0.f, 0.f, 0.f, 0.f, 0.f, 0.f, 0.f};
      c = __builtin_amdgcn_wmma_f32_16x16x4_f32(
          /*neg_a=*/false, a, /*neg_b=*/false, bv,
          /*c_mod=*/(short)0, c, /*reuse_a=*/false, /*reuse_b=*/false);

#pragma unroll
      for (int r = 0; r < 8; ++r) {
        float d = a2r[r] + b2n - 2.0f * c[r];
        // pack (ordered key, col index): min picks smallest d, ties -> lowest col
        unsigned long long pk = ((unsigned long long)f2ord(d) << 32) | (unsigned)n;
        rmin[r] = pk < rmin[r] ? pk : rmin[r];
      }
    }

    // reduce row minima across the 16 lanes of each half; exclusive owner -> plain store
#pragma unroll
    for (int r = 0; r < 8; ++r) {
      unsigned long long v = rmin[r];
#pragma unroll
      for (int m = 8; m >= 1; m >>= 1) {
        unsigned lo = (unsigned)v, hi = (unsigned)(v >> 32);
        unsigned lo2 = __shfl_xor(lo, m, 32);
        unsigned hi2 = __shfl_xor(hi, m, 32);
        unsigned long long w = ((unsigned long long)hi2 << 32) | (unsigned long long)lo2;
        v = (w < v) ? w : v;
      }
      int rr = rbase + r;
      if (col16 == 0 && rr < nA) minArg[rr] = v;
    }
  }
}

// chamfer sums + normal-consistency from row argmin (P->G) and col min (G->P)
__global__ void k_rowcol(const unsigned long long* __restrict__ rowMinArg,
                         const unsigned long long* __restrict__ colMinArg,
                         const float* __restrict__ pn, const float* __restrict__ gn,
                         float* __restrict__ prow, float* __restrict__ pcol,
                         float* __restrict__ pnrm, int npts) {
  __shared__ float sh[256];
  int stride = blockDim.x * gridDim.x;
  float rs = 0.f, cs = 0.f, ns = 0.f;
  for (int i = blockIdx.x * blockDim.x + threadIdx.x; i < npts; i += stride) {
    unsigned long long pk = rowMinArg[i];
    float dmin = ord2f((unsigned)(pk >> 32));
    unsigned idx = (unsigned)pk;
    rs += dmin;
    cs += ord2f((unsigned)(colMinArg[i] >> 32));
    float ax = pn[3 * i], ay = pn[3 * i + 1], az = pn[3 * i + 2];
    float bx = gn[3 * idx], by = gn[3 * idx + 1], bz = gn[3 * idx + 2];
    float num = ax * bx + ay * by + az * bz;
    float na = sqrtf(ax * ax + ay * ay + az * az);
    float nb = sqrtf(bx * bx + by * by + bz * bz);
    float den = fmaxf(na, 1e-8f) * fmaxf(nb, 1e-8f);
    ns += 1.f - fabsf(num / den);
  }
  float r0 = blk_sum(rs, sh);
  float r1 = blk_sum(cs, sh);
  float r2 = blk_sum(ns, sh);
  if (threadIdx.x == 0) { prow[blockIdx.x] = r0; pcol[blockIdx.x] = r1; pnrm[blockIdx.x] = r2; }
}

// deterministic fixed-order final combine
__global__ void k_final(const float* psdf, const float* peik, const float* pcnt, int nbs,
                        const float* prow, const float* pcol, const float* pnrm, int nbr,
                        const float* pedge, const float* pnp, const float* ptot,
                        const float* pbad, int nbe,
                        float* out, int Nsdf_total, int npts) {
  float ssum = 0.f, esum = 0.f, ecnt = 0.f;
  for (int i = 0; i < nbs; ++i) { ssum += psdf[i]; esum += peik[i]; ecnt += pcnt[i]; }
  float rsum = 0.f, csum = 0.f, nsum = 0.f;
  for (int i = 0; i < nbr; ++i) { rsum += prow[i]; csum += pcol[i]; nsum += pnrm[i]; }
  float edsum = 0.f, npairs = 0.f, tot = 0.f, bad = 0.f;
  for (int i = 0; i < nbe; ++i) { edsum += pedge[i]; npairs += pnp[i]; tot += ptot[i]; bad += pbad[i]; }

  float sdf = ssum / (float)Nsdf_total;
  float eik = (ecnt > 0.f) ? esum / fmaxf(ecnt, 1.f) : 0.f;
  float ch = rsum / (float)npts + csum / (float)npts;
  float nrm = nsum / (float)npts;
  float edge = (npairs > 0.f) ? edsum / fmaxf(npairs, 1.f) : 0.f;
  float wt = (tot > 0.f) ? bad / tot : 0.f;

  out[0] = 1.0f * sdf + 0.1f * eik + 1.0f * ch + 0.5f * nrm + 0.3f * edge + 0.2f * wt;
}

// ---------------------------------------------------------------------------
extern "C" void kernel_launch(void* const* d_in, const int* in_sizes, int n_in,
                              void* d_out, int out_size, void* d_ws, size_t ws_size,
                              hipStream_t stream) {
  const float* pred_sdf = (const float*)d_in[0];
  const float* gt_sdf = (const float*)d_in[1];
  const float* pred_points = (const float*)d_in[2];
  const float* gt_points = (const float*)d_in[3];
  const float* pred_normals = (const float*)d_in[4];
  const float* gt_normals = (const float*)d_in[5];
  const float* verts = (const float*)d_in[6];
  const int* faces = (const int*)d_in[7];

  const int NSDF = 100000;  // Nsdf per reference setup_inputs
  int Btot = in_sizes[0];   // B * Nsdf
  int B = Btot / NSDF;
  if (B < 1) B = 1;
  int nsdf = Btot / B;
  int npts = in_sizes[2] / 3;
  int V = in_sizes[6] / 3;
  int Fn = in_sizes[7] / 3;
  int nEdges = 3 * Fn;
  int cap = 256;
  while (cap < 2 * nEdges) cap <<= 1;
  unsigned mask = (unsigned)(cap - 1);
  int ntiles = (npts + 15) / 16;

  // carve workspace (256B aligned regions)
  char* base = (char*)d_ws;
  size_t off = 0;
  auto carve = [&](size_t bytes) -> void* {
    void* p = base + off;
    off = (off + bytes + 255) & ~(size_t)255;
    return p;
  };
  float* a2 = (float*)carve((size_t)npts * 4);
  float* b2 = (float*)carve((size_t)npts * 4);
  unsigned long long* rowMinArg = (unsigned long long*)carve((size_t)npts * 8);
  unsigned long long* colMinArg = (unsigned long long*)carve((size_t)npts * 8);
  float* fnorm = (float*)carve((size_t)Fn * 3 * 4);
  unsigned long long* hkeys = (unsigned long long*)carve((size_t)cap * 8);
  unsigned* hcnt = (unsigned*)carve((size_t)cap * 4);
  unsigned* hfa = (unsigned*)carve((size_t)cap * 4);
  unsigned* hfb = (unsigned*)carve((size_t)cap * 4);

  const int NB_SDF = 240;
  const int NB_RC = (npts + 255) / 256;
  const int NB_E = cap / 256;
  float* psdf = (float*)carve((size_t)NB_SDF * 4);
  float* peik = (float*)carve((size_t)NB_SDF * 4);
  float* pcnt = (float*)carve((size_t)NB_SDF * 4);
  float* prow = (float*)carve((size_t)NB_RC * 4);
  float* pcol = (float*)carve((size_t)NB_RC * 4);
  float* pnrm = (float*)carve((size_t)NB_RC * 4);
  float* pedge = (float*)carve((size_t)NB_E * 4);
  float* pnp = (float*)carve((size_t)NB_E * 4);
  float* ptot = (float*)carve((size_t)NB_E * 4);
  float* pbad = (float*)carve((size_t)NB_E * 4);
  (void)ws_size;
  (void)n_in;
  (void)out_size;

  k_init<<<(cap + 255) / 256, 256, 0, stream>>>(hkeys, hcnt, cap);
  k_norms<<<(npts + 255) / 256, 256, 0, stream>>>(pred_points, gt_points, a2, b2, npts);
  k_sdf_eik<<<NB_SDF, 256, 0, stream>>>(pred_sdf, gt_sdf, B, nsdf, psdf, peik, pcnt);
  k_facenorm<<<(Fn + 255) / 256, 256, 0, stream>>>(verts, faces, fnorm, Fn);
  k_edge_insert<<<(nEdges + 255) / 256, 256, 0, stream>>>(faces, hkeys, hcnt, hfa, hfb,
                                                          Fn, V, mask);
  k_edge_eval<<<NB_E, 256, 0, stream>>>(hkeys, hcnt, hfa, hfb, fnorm,
                                        pedge, pnp, ptot, pbad, cap);
  int nnBlocks = (ntiles + 7) / 8;  // 8 waves / 256-thread block
  // pass 1: for each pred point, nearest gt point (min + argmin)
  k_nnmin<<<nnBlocks, 256, 0, stream>>>(pred_points, gt_points, a2, b2,
                                        rowMinArg, npts, npts);
  // pass 2: for each gt point, nearest pred point (min only; argmin unused)
  k_nnmin<<<nnBlocks, 256, 0, stream>>>(gt_points, pred_points, b2, a2,
                                        colMinArg, npts, npts);
  k_rowcol<<<NB_RC, 256, 0, stream>>>(rowMinArg, colMinArg, pred_normals, gt_normals,
                                      prow, pcol, pnrm, npts);
  k_final<<<1, 1, 0, stream>>>(psdf, peik, pcnt, NB_SDF,
                               prow, pcol, pnrm, NB_RC,
                               pedge, pnp, ptot, pbad, NB_E,
                               (float*)d_out, Btot, npts);
}